// ChainGraphDQN_10788957847557
// MI455X (gfx1250) — compile-verified
//
#include <hip/hip_runtime.h>

// ---------------------------------------------------------------------------
// CDNA5 (gfx1250) fp32 WMMA helpers: D = A(16x4) * B(4x16) + C(16x16), wave32.
// A frag: lanes 0-15 -> M=lane, regs K=0,1 ; lanes 16-31 -> M=lane-16, K=2,3
// B frag: lanes 0-15 -> N=lane, regs K=0,1 ; lanes 16-31 -> N=lane-16, K=2,3
// C frag: reg j, lanes 0-15 -> (M=j,   N=lane)
//                 lanes16-31 -> (M=j+8, N=lane-16)
// ---------------------------------------------------------------------------
typedef float v2f __attribute__((ext_vector_type(2)));
typedef float v8f __attribute__((ext_vector_type(8)));

__device__ __forceinline__ v8f wmma4(v2f a, v2f b, v8f c) {
  // 8-arg pattern: (neg_a, A, neg_b, B, c_mod, C, reuse_a, reuse_b)
  return __builtin_amdgcn_wmma_f32_16x16x4_f32(false, a, false, b, (short)0, c,
                                               false, false);
}
__device__ __forceinline__ v8f zero8() {
  v8f z = {0.f, 0.f, 0.f, 0.f, 0.f, 0.f, 0.f, 0.f};
  return z;
}

// ---------------------------------------------------------------------------
// Degree / normalization
// ---------------------------------------------------------------------------
__global__ __launch_bounds__(256) void k_init_deg(unsigned* __restrict__ deg, int n) {
  int i = blockIdx.x * 256 + threadIdx.x;
  if (i < n) deg[i] = 1u;  // self loop
}

__global__ __launch_bounds__(256) void k_count_deg(const int* __restrict__ dst,
                                                   unsigned* __restrict__ deg, int nE) {
  int i = blockIdx.x * 256 + threadIdx.x;
  if (i < nE) atomicAdd(&deg[dst[i]], 1u);
}

__global__ __launch_bounds__(256) void k_dinv(const unsigned* __restrict__ deg,
                                              float* __restrict__ dinv, int n) {
  int i = blockIdx.x * 256 + threadIdx.x;
  if (i < n) dinv[i] = rsqrtf((float)deg[i]);
}

__global__ __launch_bounds__(256) void k_zero(float* __restrict__ p, int n) {
  int i = blockIdx.x * 256 + threadIdx.x;
  if (i < n) p[i] = 0.f;
}

// ---------------------------------------------------------------------------
// y = dinv[node] * (x @ W1)   with F_in=4, F_out=16  (one WMMA per 16 nodes)
// ---------------------------------------------------------------------------
__global__ __launch_bounds__(256) void k_xw1(const float* __restrict__ x,
                                             const float* __restrict__ W,
                                             const float* __restrict__ dinv,
                                             float* __restrict__ y, int nTiles) {
  int lane = threadIdx.x & 31, wave = threadIdx.x >> 5;
  int tile = blockIdx.x * 8 + wave;
  if (tile >= nTiles) return;  // wave-uniform
  int col = lane & 15, hi = lane >> 4, koff = hi * 2;
  int base = tile << 4;
  const float* xr = x + (size_t)(base + col) * 4;
  v2f a; a[0] = xr[koff]; a[1] = xr[koff + 1];
  v2f b; b[0] = W[koff * 16 + col]; b[1] = W[(koff + 1) * 16 + col];
  v8f c = zero8();
  c = wmma4(a, b, c);
#pragma unroll
  for (int j = 0; j < 8; ++j) {
    int node = base + j + hi * 8;
    y[(size_t)node * 16 + col] = c[j] * dinv[node];
  }
}

// y = dinv[node] * (h @ W2)   with F_in=16, F_out=16  (4 chained WMMAs)
__global__ __launch_bounds__(256) void k_xw2(const float* __restrict__ h,
                                             const float* __restrict__ W,
                                             const float* __restrict__ dinv,
                                             float* __restrict__ y, int nTiles) {
  int lane = threadIdx.x & 31, wave = threadIdx.x >> 5;
  int tile = blockIdx.x * 8 + wave;
  if (tile >= nTiles) return;
  int col = lane & 15, hi = lane >> 4, koff = hi * 2;
  int base = tile << 4;
  const float* hr = h + (size_t)(base + col) * 16;
  v8f c = zero8();
#pragma unroll
  for (int kc = 0; kc < 4; ++kc) {
    v2f a; a[0] = hr[kc * 4 + koff]; a[1] = hr[kc * 4 + koff + 1];
    v2f b; b[0] = W[(kc * 4 + koff) * 16 + col];
    b[1] = W[(kc * 4 + koff + 1) * 16 + col];
    c = wmma4(a, b, c);
  }
#pragma unroll
  for (int j = 0; j < 8; ++j) {
    int node = base + j + hi * 8;
    y[(size_t)node * 16 + col] = c[j] * dinv[node];
  }
}

// ---------------------------------------------------------------------------
// Edge scatter: acc[dst][:] += y[src][:]   (16 features, f32 atomics in L2)
// ---------------------------------------------------------------------------
__global__ __launch_bounds__(256) void k_scatter(const int* __restrict__ src,
                                                 const int* __restrict__ dst,
                                                 const float* __restrict__ y,
                                                 float* __restrict__ acc, int nE) {
  int e = blockIdx.x * 256 + threadIdx.x;
  if (e >= nE) return;
  int s = src[e], d = dst[e];
  const float4* ys = reinterpret_cast<const float4*>(y + (size_t)s * 16);
  float* ad = acc + (size_t)d * 16;
#pragma unroll
  for (int q = 0; q < 4; ++q) {
    float4 v = ys[q];
    atomicAdd(ad + q * 4 + 0, v.x);
    atomicAdd(ad + q * 4 + 1, v.y);
    atomicAdd(ad + q * 4 + 2, v.z);
    atomicAdd(ad + q * 4 + 3, v.w);
  }
}

// h = [relu]( dinv[node] * (acc + y) + bias[f] )   (self-loop folded via +y)
__global__ __launch_bounds__(256) void k_finish(const float* __restrict__ acc,
                                                const float* __restrict__ y,
                                                const float* __restrict__ dinv,
                                                const float* __restrict__ bias,
                                                float* __restrict__ h, int total,
                                                int doRelu) {
  int i = blockIdx.x * 256 + threadIdx.x;
  if (i >= total) return;
  int node = i >> 4, f = i & 15;
  float v = dinv[node] * (acc[i] + y[i]) + bias[f];
  if (doRelu) v = fmaxf(v, 0.f);
  h[i] = v;
}

// ---------------------------------------------------------------------------
// Fused per-node MLP 16->64 (relu) ->128 (relu) + per-graph mean pool.
// One wave == one graph (32 nodes == 2 M-tiles). Wf2 staged in LDS per block.
// Layer-1 C-layout is transposed to A-layout through padded LDS (stride 68).
// ---------------------------------------------------------------------------
__global__ __launch_bounds__(256) void k_mlp_pool(const float* __restrict__ h2,
                                                  const float* __restrict__ Wf1,
                                                  const float* __restrict__ bf1,
                                                  const float* __restrict__ Wf2,
                                                  const float* __restrict__ bf2,
                                                  float* __restrict__ g, int nG) {
  __shared__ float sW[64 * 132];      // Wf2, padded row stride
  __shared__ float sH[8 * 32 * 68];   // per-wave h3 scratch, padded
  int lane = threadIdx.x & 31, wave = threadIdx.x >> 5;

  for (int i = threadIdx.x; i < 64 * 128; i += 256)
    sW[(i >> 7) * 132 + (i & 127)] = Wf2[i];

  int graph = blockIdx.x * 8 + wave;
  if (graph >= nG) graph = nG - 1;  // duplicate (identical) work keeps barriers uniform
  int col = lane & 15, hi = lane >> 4, koff = hi * 2;
  size_t nb = (size_t)graph * 32;

  // A fragments for layer 1: [2 M-tiles][4 K-chunks]
  v2f a1[2][4];
#pragma unroll
  for (int mt = 0; mt < 2; ++mt) {
    const float* r = h2 + (nb + mt * 16 + col) * 16;
#pragma unroll
    for (int kc = 0; kc < 4; ++kc) {
      a1[mt][kc][0] = r[kc * 4 + koff];
      a1[mt][kc][1] = r[kc * 4 + koff + 1];
    }
  }
  __syncthreads();  // Wf2 staged

  float* mH = sH + wave * (32 * 68);
#pragma unroll
  for (int nt = 0; nt < 4; ++nt) {
    v8f c0 = zero8(), c1 = zero8();
#pragma unroll
    for (int kc = 0; kc < 4; ++kc) {
      v2f b;
      b[0] = Wf1[(kc * 4 + koff) * 64 + nt * 16 + col];
      b[1] = Wf1[(kc * 4 + koff + 1) * 64 + nt * 16 + col];
      c0 = wmma4(a1[0][kc], b, c0);
      c1 = wmma4(a1[1][kc], b, c1);
    }
    float bias = bf1[nt * 16 + col];
#pragma unroll
    for (int j = 0; j < 8; ++j) {
      mH[(j + hi * 8) * 68 + nt * 16 + col]       = fmaxf(c0[j] + bias, 0.f);
      mH[(16 + j + hi * 8) * 68 + nt * 16 + col]  = fmaxf(c1[j] + bias, 0.f);
    }
  }
  __syncthreads();  // h3 visible

  const float* A0 = mH + (size_t)col * 68;
  const float* A1 = mH + (size_t)(16 + col) * 68;
  for (int nt2 = 0; nt2 < 8; ++nt2) {
    v8f c0 = zero8(), c1 = zero8();
#pragma unroll
    for (int kc = 0; kc < 16; ++kc) {
      int k0 = kc * 4 + koff;
      v2f a0; a0[0] = A0[k0]; a0[1] = A0[k0 + 1];
      v2f av; av[0] = A1[k0]; av[1] = A1[k0 + 1];
      v2f b;  b[0] = sW[k0 * 132 + nt2 * 16 + col];
      b[1] = sW[(k0 + 1) * 132 + nt2 * 16 + col];
      c0 = wmma4(a0, b, c0);
      c1 = wmma4(av, b, c1);
    }
    float bias = bf2[nt2 * 16 + col];
    float s = 0.f;
#pragma unroll
    for (int j = 0; j < 8; ++j)
      s += fmaxf(c0[j] + bias, 0.f) + fmaxf(c1[j] + bias, 0.f);
    s += __shfl_xor(s, 16, 32);  // combine M=0..7 with M=8..15 halves
    if (lane < 16) g[(size_t)graph * 128 + nt2 * 16 + lane] = s * (1.0f / 32.0f);
  }
}

// ---------------------------------------------------------------------------
// g2 = relu(g @ Wg + bg) via WMMA (16 graphs / wave), then 4 heads 64->5.
// ---------------------------------------------------------------------------
__global__ __launch_bounds__(256) void k_head(const float* __restrict__ g,
                                              const float* __restrict__ Wg,
                                              const float* __restrict__ bg,
                                              const float* __restrict__ Wout,
                                              const float* __restrict__ bout,
                                              float* __restrict__ out, int nG) {
  __shared__ float sG2[8 * 16 * 68];
  int lane = threadIdx.x & 31, wave = threadIdx.x >> 5;
  int tile = blockIdx.x * 8 + wave;
  int maxTile = (nG >> 4) - 1;
  if (tile > maxTile) tile = maxTile;
  int col = lane & 15, hi = lane >> 4, koff = hi * 2;
  int gbase = tile << 4;

  const float* ar = g + (size_t)(gbase + col) * 128;
  v8f c[4] = {zero8(), zero8(), zero8(), zero8()};
  for (int kc = 0; kc < 32; ++kc) {
    int k0 = kc * 4 + koff;
    v2f a; a[0] = ar[k0]; a[1] = ar[k0 + 1];
#pragma unroll
    for (int nt = 0; nt < 4; ++nt) {
      v2f b;
      b[0] = Wg[k0 * 64 + nt * 16 + col];
      b[1] = Wg[(k0 + 1) * 64 + nt * 16 + col];
      c[nt] = wmma4(a, b, c[nt]);
    }
  }
  float* mG = sG2 + wave * (16 * 68);
#pragma unroll
  for (int nt = 0; nt < 4; ++nt) {
    float bias = bg[nt * 16 + col];
#pragma unroll
    for (int j = 0; j < 8; ++j)
      mG[(j + hi * 8) * 68 + nt * 16 + col] = fmaxf(c[nt][j] + bias, 0.f);
  }
  __syncthreads();

  const float* g2 = mG + (size_t)col * 68;  // one graph's 64 features
  int gidx = gbase + col;
#pragma unroll
  for (int mm = 0; mm < 2; ++mm) {
    int m = hi * 2 + mm;  // lanes 0-15: heads 0,1 ; lanes 16-31: heads 2,3
    float acc[5];
#pragma unroll
    for (int a2 = 0; a2 < 5; ++a2) acc[a2] = bout[m * 5 + a2];
    for (int f = 0; f < 64; ++f) {
      float gv = g2[f];
#pragma unroll
      for (int a2 = 0; a2 < 5; ++a2)
        acc[a2] = fmaf(gv, Wout[(m * 64 + f) * 5 + a2], acc[a2]);
    }
#pragma unroll
    for (int a2 = 0; a2 < 5; ++a2)
      out[((size_t)gidx * 4 + m) * 5 + a2] = acc[a2];
  }
}

// ---------------------------------------------------------------------------
extern "C" void kernel_launch(void* const* d_in, const int* in_sizes, int n_in,
                              void* d_out, int out_size, void* d_ws, size_t ws_size,
                              hipStream_t stream) {
  const float* x    = (const float*)d_in[0];
  const int*   ei   = (const int*)d_in[1];
  // d_in[2] = batch (contiguous, 32 nodes/graph) -> derived, not needed
  const float* W1   = (const float*)d_in[3];
  const float* b1   = (const float*)d_in[4];
  const float* W2   = (const float*)d_in[5];
  const float* b2   = (const float*)d_in[6];
  const float* Wf1  = (const float*)d_in[7];
  const float* bf1  = (const float*)d_in[8];
  const float* Wf2  = (const float*)d_in[9];
  const float* bf2  = (const float*)d_in[10];
  const float* Wg   = (const float*)d_in[11];
  const float* bg   = (const float*)d_in[12];
  const float* Wout = (const float*)d_in[13];
  const float* bout = (const float*)d_in[14];
  float* out = (float*)d_out;

  const int N = in_sizes[0] / 4;   // 262144
  const int E = in_sizes[1] / 2;   // 4194304
  const int G = in_sizes[2] / 32;  // 8192
  const int* src = ei;
  const int* dst = ei + E;

  // Workspace carve-out (256B aligned):
  char* ws = (char*)d_ws;
  size_t off = 0;
  auto take = [&](size_t bytes) -> void* {
    void* p = ws + off;
    off += (bytes + 255) & ~(size_t)255;
    return p;
  };
  unsigned* deg  = (unsigned*)take((size_t)N * 4);
  float*    dinv = (float*)take((size_t)N * 4);
  float*    ybuf = (float*)take((size_t)N * 16 * 4);
  float*    accb = (float*)take((size_t)N * 16 * 4);
  float*    hbuf = (float*)take((size_t)N * 16 * 4);
  float*    gbuf = (float*)take((size_t)G * 128 * 4);
  (void)ws_size; (void)n_in; (void)out_size;

  const int NF   = N * 16;
  const int gN   = (N + 255) / 256;
  const int gE   = (E + 255) / 256;
  const int gNF  = (NF + 255) / 256;
  const int gXW  = (N / 16 + 7) / 8;   // 16-node tiles, 8 waves per block
  const int gMLP = (G + 7) / 8;        // 1 graph per wave
  const int gHD  = (G / 16 + 7) / 8;   // 16 graphs per wave

  // Degree + normalization
  k_init_deg<<<gN, 256, 0, stream>>>(deg, N);
  k_count_deg<<<gE, 256, 0, stream>>>(dst, deg, E);
  k_dinv<<<gN, 256, 0, stream>>>(deg, dinv, N);

  // --- GCN conv 1: h1 = relu(D^-1/2 (A+I) D^-1/2 x W1 + b1) ---
  k_zero<<<gNF, 256, 0, stream>>>(accb, NF);
  k_xw1<<<gXW, 256, 0, stream>>>(x, W1, dinv, ybuf, N / 16);
  k_scatter<<<gE, 256, 0, stream>>>(src, dst, ybuf, accb, E);
  k_finish<<<gNF, 256, 0, stream>>>(accb, ybuf, dinv, b1, hbuf, NF, 1);

  // --- GCN conv 2 (no relu) ---
  k_zero<<<gNF, 256, 0, stream>>>(accb, NF);
  k_xw2<<<gXW, 256, 0, stream>>>(hbuf, W2, dinv, ybuf, N / 16);
  k_scatter<<<gE, 256, 0, stream>>>(src, dst, ybuf, accb, E);
  k_finish<<<gNF, 256, 0, stream>>>(accb, ybuf, dinv, b2, hbuf, NF, 0);

  // --- node MLP 16->64->128 + per-graph mean pool ---
  k_mlp_pool<<<gMLP, 256, 0, stream>>>(hbuf, Wf1, bf1, Wf2, bf2, gbuf, G);

  // --- graph head: relu(g@Wg+bg) @ Wout + bout ---
  k_head<<<gHD, 256, 0, stream>>>(gbuf, Wg, bg, Wout, bout, out, G);
}